// CenterLoss_29171417874844
// MI455X (gfx1250) — compile-verified
//
#include <hip/hip_runtime.h>

// ---------------------------------------------------------------------------
// CenterLoss on MI455X (gfx1250, wave32).
//
// Algebraic simplification: one_hot masking selects a single column per row,
// so loss = mean_i ||x[i] - centers[labels[i]]||^2.  Pure gather + reduce:
// ~4.2 MB of traffic (~0.2 us at 23.3 TB/s), vs 52 GFLOP / 25.6 MB for the
// faithful distmat.  Memory-bound, so the matrix pipe is free: we compute the
// per-tile row norms as the diagonal of G = d * d^T via V_WMMA_F32_16X16X4_F32.
// ---------------------------------------------------------------------------

#define CL_BATCH 4096
#define CL_FEAT  128
#define CL_TILES (CL_BATCH / 16)   // 256 tiles of 16 rows

typedef __attribute__((ext_vector_type(2))) float f32x2;
typedef __attribute__((ext_vector_type(4))) float f32x4;
typedef __attribute__((ext_vector_type(8))) float f32x8;

// One wave (32 lanes) per 16-row tile.
// WMMA f32 16x16x4 A-layout (32-bit A, 16x4): lanes 0-15 hold M=lane with
// K={k0,k0+1} in VGPR0/1; lanes 16-31 hold M=lane-16 with K={k0+2,k0+3}.
// B (4x16) with B[k][n] = d[n][k] lands in the identical register image, so
// the same float2 feeds both operands and D accumulates G = d * d^T.
// C/D layout: VGPR r, lanes 0-15 -> D[r][lane]; lanes 16-31 -> D[8+r][lane-16].
// Diagonal: acc[r] at lane r (M=r) and at lane 24+r (M=8+r).
__global__ __launch_bounds__(32) void centerloss_tile_kernel(
    const float* __restrict__ x,        // (4096, 128) f32
    const int*   __restrict__ labels,   // (4096,) int32 (harness convention)
    const float* __restrict__ centers,  // (50000, 128) f32
    float* __restrict__ partial)        // (256,) per-tile sums
{
    const int tile = blockIdx.x;       // 0..255
    const int lane = threadIdx.x;      // 0..31 (wave32)
    const int row  = tile * 16 + (lane & 15);
    const int hi   = lane >> 4;        // 0: K pair {0,1}; 1: K pair {2,3}

    const int lab = labels[row];
    const float* __restrict__ xrow = x + (size_t)row * CL_FEAT;
    const float* __restrict__ crow = centers + (size_t)lab * CL_FEAT;

    f32x8 acc = {};
#pragma unroll
    for (int k0 = 0; k0 < CL_FEAT; k0 += 4) {
        // 16B-aligned b128 loads (row base 512B-aligned, k0 % 4 == 0).
        const f32x4 xv = *reinterpret_cast<const f32x4*>(xrow + k0);
        const f32x4 cv = *reinterpret_cast<const f32x4*>(crow + k0);
        const f32x4 d  = xv - cv;

        f32x2 ab;                       // branchless half-select (v_cndmask)
        ab.x = hi ? d.z : d.x;
        ab.y = hi ? d.w : d.y;

        // D = A * B + C ; A = B = difference tile slice -> Gram accumulate.
        acc = __builtin_amdgcn_wmma_f32_16x16x4_f32(
            /*neg_a=*/false, ab, /*neg_b=*/false, ab,
            /*c_mod=*/(short)0, acc, /*reuse_a=*/false, /*reuse_b=*/false);
    }

    // Pull the 16 diagonal entries ||d_m||^2 out of the C tile (branchless).
    float s = 0.0f;
#pragma unroll
    for (int r = 0; r < 8; ++r) {
        const float v = acc[r];
        s += ((lane == r) || (lane == 24 + r)) ? v : 0.0f;
    }

    // wave32 butterfly reduction
#pragma unroll
    for (int off = 16; off > 0; off >>= 1)
        s += __shfl_xor(s, off, 32);

    if (lane == 0)
        partial[tile] = s;
}

// Deterministic final reduction: one wave, fixed summation order, no atomics.
__global__ __launch_bounds__(32) void centerloss_finalize_kernel(
    const float* __restrict__ partial, float* __restrict__ out)
{
    const int lane = threadIdx.x;
    float s = 0.0f;
#pragma unroll
    for (int i = 0; i < CL_TILES / 32; ++i)   // 8 strided adds per lane
        s += partial[lane + 32 * i];
#pragma unroll
    for (int off = 16; off > 0; off >>= 1)
        s += __shfl_xor(s, off, 32);
    if (lane == 0)
        out[0] = s * (1.0f / (float)CL_BATCH);
}

extern "C" void kernel_launch(void* const* d_in, const int* in_sizes, int n_in,
                              void* d_out, int out_size, void* d_ws, size_t ws_size,
                              hipStream_t stream) {
    (void)in_sizes; (void)n_in; (void)out_size; (void)ws_size;
    const float* x       = (const float*)d_in[0];
    const int*   labels  = (const int*)d_in[1];
    const float* centers = (const float*)d_in[2];
    float*       out     = (float*)d_out;
    float*       partial = (float*)d_ws;      // 256 floats of scratch

    centerloss_tile_kernel<<<CL_TILES, 32, 0, stream>>>(x, labels, centers, partial);
    centerloss_finalize_kernel<<<1, 32, 0, stream>>>(partial, out);
}